// FastGCN_85856396247845
// MI455X (gfx1250) — compile-verified
//
#include <hip/hip_runtime.h>

// FastGCN forward on gfx1250 (MI455X), f32 WMMA 16x16x4 path.
//
// Roofline: ~18.4 GFLOP, ~350 MB mandatory HBM traffic (single-use A_hat
// gathers) -> ~15us floor @ 23.3 TB/s. B operands (W1, H_l, W2, H_l2 <= 4MB)
// are L2-resident. f32 WMMA keeps full reference precision while putting the
// FLOPs on the matrix pipe. This revision software-pipelines fragment loads
// against WMMA issue and widens the wave tile to 32Mx64N (8 accumulators) so
// each B fragment feeds two WMMAs.

typedef __attribute__((ext_vector_type(2))) float v2f;
typedef __attribute__((ext_vector_type(8))) float v8f;

#define MODE_ROWGATHER 0  // A[rowidx[m]*lda + k]                 (X[s0] @ W1)
#define MODE_ROWCOL    1  // A[rowidx[m]*lda + colidx[k]] * scale (A_hat[s1][:,s0] @ H_l)
#define MODE_COL       2  // A[m*lda + colidx[k]] * scale         (A_hat[:,s1] @ H_l2)
#define MODE_PLAIN     3  // A[m*lda + k]                         (Hs1 @ W2)

struct Frags {
  v2f a0, a1;          // A fragments for the two 16-row M-tiles
  v2f b0, b1, b2, b3;  // B fragments for the four 16-col N-tiles
};

__device__ __forceinline__ v8f wmma4(v2f a, v2f b, v8f c) {
  return __builtin_amdgcn_wmma_f32_16x16x4_f32(false, a, false, b, (short)0, c, false, false);
}

template <bool BIAS, bool RELU, bool SCALE>
__device__ __forceinline__ void store_16x16_frag(v8f acc, float* __restrict__ C, int ldc,
                                                 int row0, int col,
                                                 const float* __restrict__ bias, float scale) {
  // C/D layout (ISA 7.12.2): VGPR r -> M = row0 + r (row0 includes +8 for
  // lanes 16..31); N = lane&15 striped across lanes.
  float bv = BIAS ? bias[col] : 0.0f;
#pragma unroll
  for (int r = 0; r < 8; ++r) {
    float v = acc[r];
    if (SCALE) v *= scale;
    if (BIAS) v += bv;
    if (RELU) v = fmaxf(v, 0.0f);
    C[(long)(row0 + r) * ldc + col] = v;
  }
}

template <int MODE, bool BIAS, bool RELU, bool SCALE>
__global__ __launch_bounds__(128) void gemm_wmma_f32(
    const float* __restrict__ A, const float* __restrict__ B,
    const float* __restrict__ bias, float* __restrict__ C,
    const int* __restrict__ rowidx, const int* __restrict__ colidx,
    int M, int N, int K, int lda, int ldb, int ldc, float scale) {
  extern __shared__ int s_col[];
  const int tid = threadIdx.x;

  // Stage the gathered-column index vector (<=16KB) into LDS once per block:
  // hot-loop index reads become ds_load_b64 instead of global gathers.
  if (MODE == MODE_ROWCOL || MODE == MODE_COL) {
    for (int i = tid; i < K; i += 128) s_col[i] = colidx[i];
    __syncthreads();
  }

  const int lane = tid & 31;
  const int wave = tid >> 5;
  const int l15  = lane & 15;
  const int hi   = lane >> 4;                 // 0: K=k,k+1 half  1: K=k+2,k+3 half
  const int m0   = (blockIdx.x * 4 + wave) * 32;  // 32 rows per wave, 128 per block
  const int n0   = blockIdx.y * 64;

  // Per-lane A row offsets in elements; max ~2.7e8 so 32-bit is safe. Keeping
  // per-lane offsets 32-bit + uniform base pointers steers codegen to
  // SADDR+VOFFSET global loads with SALU base stepping.
  int arow0, arow1;
  if (MODE == MODE_ROWGATHER || MODE == MODE_ROWCOL) {
    arow0 = rowidx[m0 + l15] * lda;
    arow1 = rowidx[m0 + 16 + l15] * lda;
  } else {
    arow0 = (m0 + l15) * lda;
    arow1 = (m0 + 16 + l15) * lda;
  }
  const int aoff0 = arow0 + hi * 2;  // used by contiguous-K modes
  const int aoff1 = arow1 + hi * 2;
  const int boff  = hi * 2 * ldb + n0 + l15;  // per-lane B element offset

  v8f acc00 = {}, acc01 = {}, acc02 = {}, acc03 = {};
  v8f acc10 = {}, acc11 = {}, acc12 = {}, acc13 = {};

  auto loadf = [&](Frags& f, int k) {
    if (MODE == MODE_ROWCOL || MODE == MODE_COL) {
      const int kb = k + hi * 2;
      const int c0 = s_col[kb];
      const int c1 = s_col[kb + 1];
      // A_hat is single-use: non-temporal so it doesn't evict L2-resident B.
      f.a0.x = __builtin_nontemporal_load(A + arow0 + c0);
      f.a0.y = __builtin_nontemporal_load(A + arow0 + c1);
      f.a1.x = __builtin_nontemporal_load(A + arow1 + c0);
      f.a1.y = __builtin_nontemporal_load(A + arow1 + c1);
    } else {
      const float* __restrict__ Ak = A + k;  // uniform base
      f.a0.x = Ak[aoff0];
      f.a0.y = Ak[aoff0 + 1];
      f.a1.x = Ak[aoff1];
      f.a1.y = Ak[aoff1 + 1];
    }
    const float* __restrict__ Bk = B + (size_t)k * ldb;  // uniform base
    f.b0.x = Bk[boff];
    f.b0.y = Bk[boff + ldb];
    f.b1.x = Bk[boff + 16];
    f.b1.y = Bk[boff + ldb + 16];
    f.b2.x = Bk[boff + 32];
    f.b2.y = Bk[boff + ldb + 32];
    f.b3.x = Bk[boff + 48];
    f.b3.y = Bk[boff + ldb + 48];
  };

  auto mm = [&](const Frags& f) {
    acc00 = wmma4(f.a0, f.b0, acc00);
    acc01 = wmma4(f.a0, f.b1, acc01);
    acc02 = wmma4(f.a0, f.b2, acc02);
    acc03 = wmma4(f.a0, f.b3, acc03);
    acc10 = wmma4(f.a1, f.b0, acc10);
    acc11 = wmma4(f.a1, f.b1, acc11);
    acc12 = wmma4(f.a1, f.b2, acc12);
    acc13 = wmma4(f.a1, f.b3, acc13);
  };

  // Two-stage software pipeline; K is always a multiple of 8 here
  // (512/4096/256/4096), so no guards are needed and EXEC stays all-1s.
  Frags f0, f1;
  loadf(f0, 0);
  loadf(f1, 4);
  int k = 0;
  for (; k < K - 8; k += 8) {
    mm(f0);
    loadf(f0, k + 8);   // overlaps with the 8 WMMAs above
    mm(f1);
    loadf(f1, k + 12);
  }
  mm(f0);
  mm(f1);

  const int r0a = m0 + hi * 8;       // M-tile 0 rows
  const int r0b = m0 + 16 + hi * 8;  // M-tile 1 rows
  store_16x16_frag<BIAS, RELU, SCALE>(acc00, C, ldc, r0a, n0 + 0  + l15, bias, scale);
  store_16x16_frag<BIAS, RELU, SCALE>(acc01, C, ldc, r0a, n0 + 16 + l15, bias, scale);
  store_16x16_frag<BIAS, RELU, SCALE>(acc02, C, ldc, r0a, n0 + 32 + l15, bias, scale);
  store_16x16_frag<BIAS, RELU, SCALE>(acc03, C, ldc, r0a, n0 + 48 + l15, bias, scale);
  store_16x16_frag<BIAS, RELU, SCALE>(acc10, C, ldc, r0b, n0 + 0  + l15, bias, scale);
  store_16x16_frag<BIAS, RELU, SCALE>(acc11, C, ldc, r0b, n0 + 16 + l15, bias, scale);
  store_16x16_frag<BIAS, RELU, SCALE>(acc12, C, ldc, r0b, n0 + 32 + l15, bias, scale);
  store_16x16_frag<BIAS, RELU, SCALE>(acc13, C, ldc, r0b, n0 + 48 + l15, bias, scale);
}

extern "C" void kernel_launch(void* const* d_in, const int* in_sizes, int n_in,
                              void* d_out, int out_size, void* d_ws, size_t ws_size,
                              hipStream_t stream) {
  (void)in_sizes; (void)n_in; (void)out_size; (void)ws_size;

  const float* X     = (const float*)d_in[0];  // [N, D_IN]
  const float* A_hat = (const float*)d_in[1];  // [N, N]
  const float* W1    = (const float*)d_in[2];  // [D_IN, D_HID]
  const float* b1    = (const float*)d_in[3];  // [D_HID]
  const float* W2    = (const float*)d_in[4];  // [D_HID, D_OUT]
  const float* b2    = (const float*)d_in[5];  // [D_OUT]
  const int*   s0    = (const int*)d_in[6];    // [S]
  const int*   s1    = (const int*)d_in[7];    // [S]
  float* out = (float*)d_out;                  // [N, D_OUT]

  const int N = 16384, D_IN = 512, D_HID = 256, D_OUT = 64, S = 4096;
  const float scale = (float)N / (float)S;     // 4.0 > 0: commutes with relu

  // Workspace: H_l [S,D_HID] | Hs1 [S,D_HID] | H_l2 [S,D_OUT]  (~9.4 MB)
  float* H_l  = (float*)d_ws;
  float* Hs1  = H_l + (size_t)S * D_HID;
  float* H_l2 = Hs1 + (size_t)S * D_HID;

  dim3 blk(128);  // 4 waves; block tile = 128 rows x 64 cols

  // K1: H_l = relu(X[s0] @ W1 + b1)        M=4096 N=256 K=512
  gemm_wmma_f32<MODE_ROWGATHER, true, true, false>
      <<<dim3(S / 128, D_HID / 64), blk, 0, stream>>>(
          X, W1, b1, H_l, s0, nullptr, S, D_HID, D_IN, D_IN, D_HID, D_HID, 1.0f);

  // K2: Hs1 = relu(scale * (A_hat[s1][:,s0] @ H_l))   M=4096 N=256 K=4096
  gemm_wmma_f32<MODE_ROWCOL, false, true, true>
      <<<dim3(S / 128, D_HID / 64), blk, (size_t)S * sizeof(int), stream>>>(
          A_hat, H_l, nullptr, Hs1, s1, s0, S, D_HID, S, N, D_HID, D_HID, scale);

  // K3: H_l2 = relu(Hs1 @ W2 + b2)         M=4096 N=64 K=256
  gemm_wmma_f32<MODE_PLAIN, true, true, false>
      <<<dim3(S / 128, D_OUT / 64), blk, 0, stream>>>(
          Hs1, W2, b2, H_l2, nullptr, nullptr, S, D_OUT, D_HID, D_HID, D_OUT, D_OUT, 1.0f);

  // K4: out = scale * (A_hat[:, s1] @ H_l2)  M=16384 N=64 K=4096
  gemm_wmma_f32<MODE_COL, false, false, true>
      <<<dim3(N / 128, D_OUT / 64), blk, (size_t)S * sizeof(int), stream>>>(
          A_hat, H_l2, nullptr, out, nullptr, s1, N, D_OUT, S, N, D_OUT, D_OUT, scale);
}